// ProbAttention_43035572306466
// MI455X (gfx1250) — compile-verified
//
#include <hip/hip_runtime.h>
#include <hip/hip_bf16.h>
#include <math.h>

// ---------------- types ----------------
typedef __attribute__((ext_vector_type(2))) float v2f;
typedef __attribute__((ext_vector_type(8))) float v8f;

#define BH      32          // b*h
#define LQ      4096
#define DIM     64
#define NTOP    41
#define NTOPP   48          // padded to 3 tiles of 16
#define NCHUNK  4           // key-dim split for flash partials
#define KEYS_PER_CHUNK (LQ / NCHUNK)      // 1024
#define TILES_PER_CHUNK (KEYS_PER_CHUNK / 16) // 64
#define SAMPK   41

// ---------------- kernel 1: v mean + broadcast fill ----------------
__global__ void pa_mean_fill(const float* __restrict__ v, float* __restrict__ out) {
    const int bh = blockIdx.x;
    const int t  = threadIdx.x;              // 256 threads
    __shared__ float ssum[256];
    __shared__ float smean[DIM];

    const int d   = t & 63;
    const int seg = t >> 6;                  // 4 segments of 1024 rows
    const float* vb = v + (size_t)bh * LQ * DIM;

    float acc = 0.f;
    for (int row = seg * 1024; row < (seg + 1) * 1024; ++row)
        acc += vb[(size_t)row * DIM + d];
    ssum[t] = acc;
    __syncthreads();
    if (t < DIM)
        smean[t] = (ssum[t] + ssum[t + 64] + ssum[t + 128] + ssum[t + 192]) * (1.0f / (float)LQ);
    __syncthreads();

    float* ob = out + (size_t)bh * LQ * DIM;
    for (int i = t; i < LQ * DIM; i += 256)
        ob[i] = smean[i & 63];
}

// ---------------- kernel 2: sampled sparsity score m = max - mean ----------------
__global__ void pa_sample_score(const float* __restrict__ q, const float* __restrict__ k,
                                const int* __restrict__ idxs, float* __restrict__ wsM) {
    const int lane   = threadIdx.x & 31;
    const int waveId = blockIdx.x * (blockDim.x >> 5) + (threadIdx.x >> 5);
    const int qi = waveId & (LQ - 1);
    const int bh = waveId >> 12;

    const float* qrow = q + ((size_t)bh * LQ + qi) * DIM;
    const float qx = qrow[2 * lane];
    const float qy = qrow[2 * lane + 1];

    float mx = -1e30f, sm = 0.f;
    for (int s = 0; s < SAMPK; ++s) {
        const int idx = idxs[qi * SAMPK + s];
        const float* kr = k + ((size_t)bh * LQ + idx) * DIM;
        float part = qx * kr[2 * lane] + qy * kr[2 * lane + 1];
        #pragma unroll
        for (int off = 1; off < 32; off <<= 1)
            part += __shfl_xor(part, off);
        mx = fmaxf(mx, part);
        sm += part;
    }
    if (lane == 0)
        wsM[bh * LQ + qi] = mx - sm * (1.0f / (float)SAMPK);
}

// ---------------- kernel 3: top-41 per (b,h) ----------------
__global__ void pa_topk(const float* __restrict__ wsM, int* __restrict__ wsTop) {
    const int bh = blockIdx.x;
    const int t  = threadIdx.x;              // 256 threads
    __shared__ float vals[LQ];               // 16 KB
    __shared__ float rv[256];
    __shared__ int   ri[256];

    for (int i = t; i < LQ; i += 256) vals[i] = wsM[bh * LQ + i];
    __syncthreads();

    for (int it = 0; it < NTOP; ++it) {
        float best = -1e30f; int bi = LQ;
        for (int i = t; i < LQ; i += 256) {
            float vv = vals[i];
            if (vv > best) { best = vv; bi = i; }
        }
        rv[t] = best; ri[t] = bi;
        __syncthreads();
        for (int s = 128; s > 0; s >>= 1) {
            if (t < s) {
                if (rv[t + s] > rv[t] || (rv[t + s] == rv[t] && ri[t + s] < ri[t])) {
                    rv[t] = rv[t + s]; ri[t] = ri[t + s];
                }
            }
            __syncthreads();
        }
        if (t == 0) { wsTop[bh * NTOPP + it] = ri[0]; vals[ri[0]] = -3e38f; }
        __syncthreads();
    }
}

// ---------------- kernel 4: flash attention partials for top queries (WMMA) ----------------
// grid = (NCHUNK, BH), block = 96 (3 waves; wave w handles query rows [16w,16w+16))
__global__ void __launch_bounds__(96)
pa_flash_partial(const float* __restrict__ q, const float* __restrict__ k,
                 const float* __restrict__ v, const int* __restrict__ wsTop,
                 float* __restrict__ wsPm, float* __restrict__ wsPl,
                 float* __restrict__ wsPo) {
    const int chunk = blockIdx.x;
    const int bh    = blockIdx.y;
    const int wave  = threadIdx.x >> 5;
    const int lane  = threadIdx.x & 31;
    const int half  = lane >> 4;             // 0: rows/K low, 1: high
    const int col   = lane & 15;

    __shared__ int   sTop[NTOPP];
    __shared__ float sP[3][16][17];          // per-wave P tile, C->A relayout

    if (threadIdx.x < NTOPP)
        sTop[threadIdx.x] = (threadIdx.x < NTOP) ? wsTop[bh * NTOPP + threadIdx.x] : 0;
    __syncthreads();

    const int qBase   = wave * 16;
    const int myRowA  = qBase + col;         // A-matrix row m = lane&15
    const bool rowOk  = (myRowA < NTOP);
    const float scale = 0.125f;              // 1/sqrt(64)

    // Q fragments in A layout: lane holds A[m][k0], A[m][k0+1], k0 = 4*ks + 2*half
    v2f qf[16];
    {
        const float* qrow = q + ((size_t)bh * LQ + sTop[myRowA]) * DIM;
        #pragma unroll
        for (int ks = 0; ks < 16; ++ks) {
            const int k0 = ks * 4 + 2 * half;
            if (rowOk) { qf[ks].x = qrow[k0] * scale; qf[ks].y = qrow[k0 + 1] * scale; }
            else       { qf[ks].x = 0.f;              qf[ks].y = 0.f; }
        }
    }

    float mrun[8], lrun[8];
    #pragma unroll
    for (int r = 0; r < 8; ++r) { mrun[r] = -1e30f; lrun[r] = 0.f; }
    v8f o[4] = {v8f{}, v8f{}, v8f{}, v8f{}};

    for (int t = 0; t < TILES_PER_CHUNK; ++t) {
        const int kt = (chunk * TILES_PER_CHUNK + t) * 16;  // first key of tile
        const float* kb = k + ((size_t)bh * LQ + kt) * DIM;

        // S = (Q*scale) @ K^T  via 16 x WMMA f32 16x16x4
        v8f s = v8f{};
        #pragma unroll
        for (int ks = 0; ks < 16; ++ks) {
            const int k0 = ks * 4 + 2 * half;
            v2f b;      // B[k][n] = K[kt+n][k]
            b.x = kb[(size_t)col * DIM + k0];
            b.y = kb[(size_t)col * DIM + k0 + 1];
            s = __builtin_amdgcn_wmma_f32_16x16x4_f32(false, qf[ks], false, b,
                                                      (short)0, s, false, false);
        }

        // online softmax update (row r lives in lanes of this half; N across 16 lanes)
        #pragma unroll
        for (int r = 0; r < 8; ++r) {
            float v0 = s[r];
            float tm = v0;
            #pragma unroll
            for (int off = 1; off <= 8; off <<= 1)
                tm = fmaxf(tm, __shfl_xor(tm, off));
            const float mn    = fmaxf(mrun[r], tm);
            const float alpha = __expf(mrun[r] - mn);
            const float p     = __expf(v0 - mn);
            float ts = p;
            #pragma unroll
            for (int off = 1; off <= 8; off <<= 1)
                ts += __shfl_xor(ts, off);
            lrun[r] = lrun[r] * alpha + ts;
            mrun[r] = mn;
            #pragma unroll
            for (int n = 0; n < 4; ++n) o[n][r] *= alpha;
            s[r] = p;
        }

        // C->A relayout of P through LDS
        __syncthreads();
        #pragma unroll
        for (int r = 0; r < 8; ++r)
            sP[wave][r + 8 * half][col] = s[r];
        __syncthreads();

        // O += P @ V_tile  via 16 x WMMA f32 16x16x4 (K = 16 keys, 4 N-tiles)
        const float* vb = v + ((size_t)bh * LQ + kt) * DIM;
        #pragma unroll
        for (int ks2 = 0; ks2 < 4; ++ks2) {
            const int kk0 = ks2 * 4 + 2 * half;
            v2f a;   // A[m][k] = P[col][kk0..kk0+1]
            a.x = sP[wave][col][kk0];
            a.y = sP[wave][col][kk0 + 1];
            #pragma unroll
            for (int n = 0; n < 4; ++n) {
                const int ncol = n * 16 + col;
                v2f b;   // B[k][n] = V[kt+kk0+k][ncol]
                b.x = vb[(size_t)kk0 * DIM + ncol];
                b.y = vb[(size_t)(kk0 + 1) * DIM + ncol];
                o[n] = __builtin_amdgcn_wmma_f32_16x16x4_f32(false, a, false, b,
                                                             (short)0, o[n], false, false);
            }
        }
        __syncthreads();
    }

    // write partials: rows 0..47, stats once per row (col==0 lane of each half)
    const int base = (bh * NCHUNK + chunk) * NTOPP;
    #pragma unroll
    for (int r = 0; r < 8; ++r) {
        const int row = qBase + r + 8 * half;
        if (col == 0) { wsPm[base + row] = mrun[r]; wsPl[base + row] = lrun[r]; }
        #pragma unroll
        for (int n = 0; n < 4; ++n)
            wsPo[(size_t)(base + row) * DIM + n * 16 + col] = o[n][r];
    }
}

// ---------------- kernel 5: combine chunk partials, scatter into output ----------------
__global__ void pa_combine_scatter(const int* __restrict__ wsTop,
                                   const float* __restrict__ wsPm,
                                   const float* __restrict__ wsPl,
                                   const float* __restrict__ wsPo,
                                   float* __restrict__ out) {
    const int bh = blockIdx.x;
    for (int i = threadIdx.x; i < NTOP * DIM; i += blockDim.x) {
        const int row = i >> 6;
        const int d   = i & 63;
        const int pb  = bh * NCHUNK * NTOPP;

        float mstar = -1e30f;
        #pragma unroll
        for (int c = 0; c < NCHUNK; ++c)
            mstar = fmaxf(mstar, wsPm[pb + c * NTOPP + row]);

        float lsum = 0.f, osum = 0.f;
        #pragma unroll
        for (int c = 0; c < NCHUNK; ++c) {
            const int   pi = pb + c * NTOPP + row;
            const float sc = __expf(wsPm[pi] - mstar);
            lsum += wsPl[pi] * sc;
            osum += wsPo[(size_t)pi * DIM + d] * sc;
        }
        const int qi = wsTop[bh * NTOPP + row];
        out[((size_t)bh * LQ + qi) * DIM + d] = osum / lsum;
    }
}

// ---------------- launcher ----------------
extern "C" void kernel_launch(void* const* d_in, const int* in_sizes, int n_in,
                              void* d_out, int out_size, void* d_ws, size_t ws_size,
                              hipStream_t stream) {
    const float* q    = (const float*)d_in[0];
    const float* k    = (const float*)d_in[1];
    const float* v    = (const float*)d_in[2];
    const int*   idxs = (const int*)d_in[3];
    float*       out  = (float*)d_out;

    // workspace layout (floats / ints)
    float* wsF  = (float*)d_ws;
    float* wsM  = wsF;                                   // BH*LQ           = 131072 f
    int*   wsTop = (int*)(wsF + BH * LQ);                // BH*NTOPP        = 1536  i
    float* wsPm = wsF + BH * LQ + BH * NTOPP;            // BH*NCHUNK*NTOPP = 6144  f
    float* wsPl = wsPm + BH * NCHUNK * NTOPP;            // 6144 f
    float* wsPo = wsPl + BH * NCHUNK * NTOPP;            // BH*NCHUNK*NTOPP*DIM = 393216 f

    // 1) fill output with per-head mean of v
    pa_mean_fill<<<BH, 256, 0, stream>>>(v, out);

    // 2) sampled sparsity scores (one wave per query; 8 waves / block)
    pa_sample_score<<<(BH * LQ) / 8, 256, 0, stream>>>(q, k, idxs, wsM);

    // 3) top-41 queries per head
    pa_topk<<<BH, 256, 0, stream>>>(wsM, wsTop);

    // 4) flash attention partials over 4 key chunks (WMMA f32 16x16x4)
    pa_flash_partial<<<dim3(NCHUNK, BH), 96, 0, stream>>>(q, k, v, wsTop, wsPm, wsPl, wsPo);

    // 5) merge partials, scatter into output rows
    pa_combine_scatter<<<BH, 256, 0, stream>>>(wsTop, wsPm, wsPl, wsPo, out);
}